// LocalAttention_55052890800496
// MI455X (gfx1250) — compile-verified
//
#include <hip/hip_runtime.h>

typedef __attribute__((ext_vector_type(2))) float v2f;
typedef __attribute__((ext_vector_type(8))) float v8f;

namespace {
constexpr int B = 4, L = 2048, H = 16, E = 64, D = 64;
constexpr int HALFW = 32;              // window/2
constexpr int WAVES = 2;               // q-tiles (waves) per block
constexpr int TPB = 32 * WAVES;        // 64 threads
constexpr int QROWS = 16 * WAVES;      // 32 query rows per block
constexpr int NK = QROWS + 64;         // staged key rows: [qb-32, qb+64) -> 96
constexpr int KSTR = 68;               // LDS row stride (floats) for K panel
constexpr int VSTR = 72;               // LDS row stride (floats) for V panel
constexpr int ASTR = 84;               // LDS row stride (floats) for A tile
constexpr int VOFF = NK * KSTR;        // V panel offset (floats) = 6528
constexpr int SMEMF = NK * KSTR + NK * VSTR;  // 13440 floats = 53760 B
constexpr int APW = 1680;              // per-wave A slice (>= 16*84), aliases K panel
}

__global__ __launch_bounds__(TPB) void local_attn_wmma_f32(
    const float* __restrict__ Q, const float* __restrict__ K,
    const float* __restrict__ V, float* __restrict__ O)
{
  __shared__ float smem[SMEMF];

  const int tid  = threadIdx.x;
  const int wave = tid >> 5;
  const int lane = tid & 31;
  const int lp   = lane & 15;
  const int hi   = lane >> 4;           // half-wave select (K-pair / row+8)

  const int qb = blockIdx.x * QROWS;    // first query row of this block
  const int h  = blockIdx.y;
  const int b  = blockIdx.z;

  const size_t rowstr = (size_t)H * E;  // 1024 floats between consecutive l
  const float* Qbh = Q + ((size_t)b * L) * rowstr + (size_t)h * E;
  const float* Kbh = K + ((size_t)b * L) * rowstr + (size_t)h * E;
  const float* Vbh = V + ((size_t)b * L) * rowstr + (size_t)h * D;
  float*       Obh = const_cast<float*>(O) + ((size_t)b * L) * rowstr + (size_t)h * D;

  // ---- Stage K and V panels (96 rows x 64 floats) into LDS, coalesced float4 ----
  for (int i = tid; i < NK * (E / 4); i += TPB) {
    const int row = i >> 4;             // 0..95
    const int c4  = i & 15;             // float4 index within row
    int gr = qb - HALFW + row;
    gr = gr < 0 ? 0 : (gr >= L ? L - 1 : gr);   // clamp; masked later
    const float4 kd = *(const float4*)(Kbh + (size_t)gr * rowstr + c4 * 4);
    const float4 vd = *(const float4*)(Vbh + (size_t)gr * rowstr + c4 * 4);
    *(float4*)(&smem[row * KSTR + c4 * 4]) = kd;
    *(float4*)(&smem[VOFF + row * VSTR + c4 * 4]) = vd;
  }
  __syncthreads();

  // ---- Load Q A-fragments straight from global ----
  // A-matrix 16x4 fp32 layout: lanes 0-15 hold K={0,1}, lanes 16-31 K={2,3}.
  const int q0 = qb + wave * 16;
  v2f qa[16];
  const float* qrow = Qbh + (size_t)(q0 + lp) * rowstr;
  #pragma unroll
  for (int kk = 0; kk < 16; ++kk)
    qa[kk] = *(const v2f*)(qrow + 4 * kk + 2 * hi);

  // ---- S = Q * K^T over the 80-key band: 5 tiles x 16 fp32-WMMA steps ----
  v8f acc[5];
  #pragma unroll
  for (int t = 0; t < 5; ++t) {
    v8f a = {};
    const float* krow = &smem[(wave * 16 + t * 16 + lp) * KSTR + 2 * hi];
    #pragma unroll
    for (int kk = 0; kk < 16; ++kk) {
      v2f kb = *(const v2f*)(krow + 4 * kk);
      a = __builtin_amdgcn_wmma_f32_16x16x4_f32(false, qa[kk], false, kb,
                                                (short)0, a, false, false);
    }
    acc[t] = a;
  }

  // ---- Band mask + row softmax (C/D layout: vgpr r -> row r+8*hi, lane&15 -> col) ----
  const float scale = 0.125f;           // 1/sqrt(64)
  #pragma unroll
  for (int r = 0; r < 8; ++r) {
    const int i = q0 + r + 8 * hi;      // query row
    float m = -1e30f;
    #pragma unroll
    for (int t = 0; t < 5; ++t) {
      const int j = q0 - HALFW + 16 * t + lp;   // key column
      const bool ok = (j >= i - HALFW) && (j < i + HALFW) && (j >= 0) && (j < L);
      const float x = ok ? acc[t][r] * scale : -1e30f;
      acc[t][r] = x;
      m = fmaxf(m, x);
    }
    m = fmaxf(m, __shfl_xor(m, 1, 32));
    m = fmaxf(m, __shfl_xor(m, 2, 32));
    m = fmaxf(m, __shfl_xor(m, 4, 32));
    m = fmaxf(m, __shfl_xor(m, 8, 32));
    float s = 0.0f;
    #pragma unroll
    for (int t = 0; t < 5; ++t) {
      const float p = __expf(acc[t][r] - m);
      acc[t][r] = p;
      s += p;
    }
    s += __shfl_xor(s, 1, 32);
    s += __shfl_xor(s, 2, 32);
    s += __shfl_xor(s, 4, 32);
    s += __shfl_xor(s, 8, 32);
    const float inv = 1.0f / s;
    #pragma unroll
    for (int t = 0; t < 5; ++t) acc[t][r] *= inv;
  }

  // ---- Re-layout A via LDS (alias the consumed K panel; barrier first) ----
  __syncthreads();                      // everyone is done reading the K panel
  float* aw = &smem[wave * APW];
  #pragma unroll
  for (int t = 0; t < 5; ++t) {
    #pragma unroll
    for (int r = 0; r < 8; ++r)
      aw[(r + 8 * hi) * ASTR + 16 * t + lp] = acc[t][r];
  }

  // ---- O = A * V : 4 output d-tiles x 20 fp32-WMMA K-steps ----
  #pragma unroll
  for (int nt = 0; nt < 4; ++nt) {
    v8f o = {};
    #pragma unroll
    for (int c = 0; c < 20; ++c) {
      const v2f af = *(const v2f*)(aw + lp * ASTR + 4 * c + 2 * hi);
      const int vr = wave * 16 + 4 * c + 2 * hi;
      v2f vf;
      vf.x = smem[VOFF + vr * VSTR + 16 * nt + lp];
      vf.y = smem[VOFF + (vr + 1) * VSTR + 16 * nt + lp];
      o = __builtin_amdgcn_wmma_f32_16x16x4_f32(false, af, false, vf,
                                                (short)0, o, false, false);
    }
    #pragma unroll
    for (int r = 0; r < 8; ++r)
      Obh[(size_t)(q0 + r + 8 * hi) * rowstr + 16 * nt + lp] = o[r];
  }
}

extern "C" void kernel_launch(void* const* d_in, const int* in_sizes, int n_in,
                              void* d_out, int out_size, void* d_ws, size_t ws_size,
                              hipStream_t stream) {
  (void)in_sizes; (void)n_in; (void)out_size; (void)d_ws; (void)ws_size;
  const float* q = (const float*)d_in[0];
  const float* k = (const float*)d_in[1];
  const float* v = (const float*)d_in[2];
  float* out = (float*)d_out;
  dim3 grid(L / QROWS, H, B);           // (64, 16, 4)
  dim3 block(TPB);                      // 64 threads = 2 wave32
  local_attn_wmma_f32<<<grid, block, 0, stream>>>(q, k, v, out);
}